// HolographicMemory_12463995093831
// MI455X (gfx1250) — compile-verified
//
#include <hip/hip_runtime.h>
#include <math.h>

// ---------------------------------------------------------------------------
// HolographicMemory retrieval for MI455X (gfx1250, wave32)
//   d_in[0]: cue    [D=2048] fp32
//   d_in[1]: memory [C=131072, D] fp32
//   d_in[2]: k (ignored -- only top-1 outputs are returned)
//   d_out  : [D retrieved row][1 top sim][1 confidence]  (2050 fp32)
// Bandwidth-bound GEMV: WMMA f32 16x16x4 for dots (cue in B column 0),
// co-executing VALU FMAs for row norms, async global->LDS double buffering
// via inline-asm global_load_async_to_lds_b128 + s_wait_asynccnt.
// ---------------------------------------------------------------------------

typedef float v2f __attribute__((ext_vector_type(2)));
typedef float v4f __attribute__((ext_vector_type(4)));
typedef float v8f __attribute__((ext_vector_type(8)));

// Flip to 0 if the assembler rejects the async mnemonic (sync LDS fallback).
#define USE_ASYNC_ASM 1

// Generic (flat) shared pointer -> raw 32-bit LDS byte offset.
__device__ __forceinline__ unsigned lds_offset(const void* p) {
  return (unsigned)(size_t)(__attribute__((address_space(3))) const void*)p;
}

__device__ __forceinline__ void async_cp16(const float* g, float* l) {
#if USE_ASYNC_ASM
  const unsigned  laddr = lds_offset(l);
  const unsigned long long gaddr = (unsigned long long)(size_t)g;
  // VDST = LDS byte address, VADDR = 64-bit global address, SADDR = off.
  asm volatile("global_load_async_to_lds_b128 %0, %1, off"
               :
               : "v"(laddr), "v"(gaddr)
               : "memory");
#else
  *(v4f*)l = *(const v4f*)g;   // synchronous fallback
#endif
}

template <int N>
__device__ __forceinline__ void wait_async_le() {
#if USE_ASYNC_ASM
  asm volatile("s_wait_asynccnt %0" :: "i"(N) : "memory");
#endif
}

// Tile: 16 rows x 32 cols fp32, LDS row stride 36 floats (144B) so the WMMA
// A-fragment ds_load_b64 pattern (lane*144 + 8*half) is 64-bank conflict-free.
#define T_ROWS   16
#define T_COLS   32
#define T_STRIDE 36
#define T_FLOATS (T_ROWS * T_STRIDE)   // 576
#define WAVES_PER_BLOCK 8
#define ROWS_PER_BLOCK  (WAVES_PER_BLOCK * T_ROWS)  // 128
#define MAX_D 2048

// One stage = 4 async b128 instructions per wave: lane covers
// (row = i*4 + lane/8, 16B chunk = lane%8); 2 KiB moved per stage.
__device__ __forceinline__ void stage_tile(const float* gRow0, int D, float* ldsBuf, int lane) {
  const int r0  = lane >> 3;   // 0..3
  const int c16 = lane & 7;    // 0..7 (16B units)
  #pragma unroll
  for (int i = 0; i < 4; ++i) {
    const int row = i * 4 + r0;
    async_cp16(gRow0 + (size_t)row * D + c16 * 4, ldsBuf + row * T_STRIDE + c16 * 4);
  }
}

__global__ void __launch_bounds__(256)
hm_gemv_wmma(const float* __restrict__ cue, const float* __restrict__ mem,
             float* __restrict__ dots, float* __restrict__ nrms, int D) {
  __shared__ float cueLds[MAX_D];
  __shared__ float tileLds[WAVES_PER_BLOCK * 2 * T_FLOATS];

  const int tid  = threadIdx.x;
  const int lane = tid & 31;
  const int wave = tid >> 5;

  for (int i = tid; i < D; i += 256) cueLds[i] = cue[i];
  __syncthreads();

  const int rowBase   = blockIdx.x * ROWS_PER_BLOCK + wave * T_ROWS;
  const float* gbase  = mem + (size_t)rowBase * D;
  float* buf0 = tileLds + wave * (2 * T_FLOATS);
  float* buf1 = buf0 + T_FLOATS;

  const int laneN   = lane & 15;            // N index within half-wave
  const int halfsel = (lane >> 4) << 1;     // K offset: 0 (lanes 0-15) or 2 (lanes 16-31)
  const int nChunks = D / T_COLS;           // 64 for D=2048

  v8f acc = {0.f, 0.f, 0.f, 0.f, 0.f, 0.f, 0.f, 0.f};
  const v2f vzero = {0.f, 0.f};
  float nrm = 0.f;

  stage_tile(gbase, D, buf0, lane);
  float* cur = buf0;
  float* nxt = buf1;

  for (int ch = 0; ch < nChunks; ++ch) {
    if (ch + 1 < nChunks) {
      stage_tile(gbase + (size_t)(ch + 1) * T_COLS, D, nxt, lane);
      wait_async_le<4>();   // previous stage's 4 async ops complete (in-order)
    } else {
      wait_async_le<0>();
    }
    const float* cuep = cueLds + ch * T_COLS;
    #pragma unroll
    for (int kk = 0; kk < T_COLS; kk += 4) {
      // A fragment (16x4 fp32 ISA layout): lanes 0-15 hold K={kk,kk+1} of row
      // laneN, lanes 16-31 hold K={kk+2,kk+3}.
      v2f a = *(const v2f*)(cur + laneN * T_STRIDE + kk + halfsel);
      // B fragment (4x16): only column N=0 nonzero = cue chunk.
      v2f bs = *(const v2f*)(cuep + kk + halfsel);
      v2f b  = (laneN == 0) ? bs : vzero;
      // Row sum-of-squares on the VALU, co-executing with the XDL WMMA.
      nrm = __builtin_fmaf(a.x, a.x, nrm);
      nrm = __builtin_fmaf(a.y, a.y, nrm);
      acc = __builtin_amdgcn_wmma_f32_16x16x4_f32(false, a, false, b,
                                                  (short)0, acc, false, false);
    }
    float* t = cur; cur = nxt; nxt = t;
  }

  // Row r sumsq = lane r partial + lane r+16 partial.
  float nrmTot = nrm + __shfl_xor(nrm, 16, 32);
  if (lane < 16) nrms[rowBase + lane] = nrmTot;

  // D-matrix column N=0: lane 0 has M=0..7 in acc[0..7], lane 16 has M=8..15.
  if (laneN == 0) {
    const int base = rowBase + ((lane >> 4) << 3);
    #pragma unroll
    for (int m = 0; m < 8; ++m) dots[base + m] = acc[m];
  }
}

// Single-block finalize: cue norm, top-2 + argmax (lowest-index tie-break,
// matching lax.top_k), confidence, and gather of the winning row.
__global__ void __launch_bounds__(1024)
hm_finalize(const float* __restrict__ cue, const float* __restrict__ mem,
            const float* __restrict__ dots, const float* __restrict__ nrms,
            float* __restrict__ out, int D, int C) {
  __shared__ float sV1[1024];
  __shared__ float sV2[1024];
  __shared__ int   sI1[1024];
  __shared__ float sCueN;
  const int tid = threadIdx.x;

  float cs = 0.f;
  for (int i = tid; i < D; i += 1024) { float v = cue[i]; cs = __builtin_fmaf(v, v, cs); }
  sV1[tid] = cs;
  __syncthreads();
  for (int off = 512; off > 0; off >>= 1) {
    if (tid < off) sV1[tid] += sV1[tid + off];
    __syncthreads();
  }
  if (tid == 0) sCueN = fmaxf(sqrtf(sV1[0]), 1e-8f);
  __syncthreads();

  float v1 = -INFINITY, v2 = -INFINITY;
  int i1 = -1;
  for (int i = tid; i < C; i += 1024) {
    const float mn = fmaxf(sqrtf(nrms[i]), 1e-8f);
    const float s  = dots[i] / mn;          // sims * cue_n (uniform scale)
    if (s > v1)      { v2 = v1; v1 = s; i1 = i; }
    else if (s > v2) { v2 = s; }
  }
  sV1[tid] = v1; sV2[tid] = v2; sI1[tid] = i1;
  __syncthreads();
  for (int off = 512; off > 0; off >>= 1) {
    if (tid < off) {
      const float av1 = sV1[tid],       av2 = sV2[tid];       const int ai1 = sI1[tid];
      const float bv1 = sV1[tid + off], bv2 = sV2[tid + off]; const int bi1 = sI1[tid + off];
      if (bv1 > av1 || (bv1 == av1 && (unsigned)bi1 < (unsigned)ai1)) {
        sV1[tid] = bv1; sI1[tid] = bi1; sV2[tid] = fmaxf(bv2, av1);
      } else {
        sV2[tid] = fmaxf(av2, bv1);
      }
    }
    __syncthreads();
  }

  if (tid == 0) {
    const float cueN = sCueN;
    const float top  = sV1[0] / cueN;
    const float sec  = sV2[0] / cueN;
    const float interf = fmaxf(sec, 0.0f);   // masked-max includes the injected 0.0
    out[D]     = top;
    out[D + 1] = top / (top + interf + 1e-9f);
  }
  __syncthreads();

  const int idx = sI1[0];
  const float* src = mem + (size_t)idx * D;
  for (int i = tid; i < D; i += 1024) out[i] = src[i];
}

extern "C" void kernel_launch(void* const* d_in, const int* in_sizes, int n_in,
                              void* d_out, int out_size, void* d_ws, size_t ws_size,
                              hipStream_t stream) {
  const float* cue = (const float*)d_in[0];
  const float* mem = (const float*)d_in[1];
  const int D = in_sizes[0];                 // 2048
  const int C = in_sizes[1] / D;             // 131072
  float* dots = (float*)d_ws;                // C floats
  float* nrms = dots + C;                    // C floats   (1 MiB total scratch)
  float* out  = (float*)d_out;

  const int blocks = C / ROWS_PER_BLOCK;     // 1024
  hm_gemv_wmma<<<blocks, 256, 0, stream>>>(cue, mem, dots, nrms, D);
  hm_finalize<<<1, 1024, 0, stream>>>(cue, mem, dots, nrms, out, D, C);
}